// FeedbackLoop_17102559773314
// MI455X (gfx1250) — compile-verified
//
#include <hip/hip_runtime.h>
#include <math.h>

typedef __attribute__((ext_vector_type(2))) float v2f;
typedef __attribute__((ext_vector_type(8))) float v8f;

#define PI_F 3.14159265358979323846f

// ---------------------------------------------------------------------------
// cross-lane helpers (wave32)
// ---------------------------------------------------------------------------
__device__ __forceinline__ float bperm(float x, int srcLane) {
  int r = __builtin_amdgcn_ds_bpermute(srcLane << 2, __builtin_bit_cast(int, x));
  return __builtin_bit_cast(float, r);
}

__device__ __forceinline__ void lds_fence() {
  asm volatile("s_wait_dscnt 0" ::: "memory");
}

// ---------------------------------------------------------------------------
// WMMA f32 16x16x4: D = A(16x4) * B(4x16) + C
// ---------------------------------------------------------------------------
__device__ __forceinline__ v8f wmma4(v2f a, v2f b, v8f c) {
  return __builtin_amdgcn_wmma_f32_16x16x4_f32(false, a, false, b, (short)0, c,
                                               false, false);
}

// cr/ci += sgn * (A @ B), complex 16x16x16.  A,B row-major (re/im planes) in LDS.
__device__ __forceinline__ void cgemm(const float* Ar, const float* Ai,
                                      const float* Br, const float* Bi,
                                      v8f& cr, v8f& ci, float sgn) {
  const int l  = threadIdx.x & 31;
  const int m  = l & 15;          // A row / B col for this lane
  const int ko = (l >> 4) << 1;   // K sub-offset per lane half (0 or 2)
#pragma unroll
  for (int k0 = 0; k0 < 16; k0 += 4) {
    v2f ar, ai, nai, br, bi;
    ar.x = sgn * Ar[m * 16 + k0 + ko];
    ar.y = sgn * Ar[m * 16 + k0 + ko + 1];
    ai.x = sgn * Ai[m * 16 + k0 + ko];
    ai.y = sgn * Ai[m * 16 + k0 + ko + 1];
    nai  = -ai;
    br.x = Br[(k0 + ko) * 16 + m];
    br.y = Br[(k0 + ko + 1) * 16 + m];
    bi.x = Bi[(k0 + ko) * 16 + m];
    bi.y = Bi[(k0 + ko + 1) * 16 + m];
    cr = wmma4(ar,  br, cr);   // + sgn*Ar*Br
    cr = wmma4(nai, bi, cr);   // - sgn*Ai*Bi
    ci = wmma4(ar,  bi, ci);   // + sgn*Ar*Bi
    ci = wmma4(ai,  br, ci);   // + sgn*Ai*Br
  }
}

// ---------------------------------------------------------------------------
// In-register 16x16 complex Gauss-Jordan inverse (no pivoting; matrix is
// D*(I - E) with ||E|| < 1 so leading minors are safe).
// Layout = WMMA accumulator layout: lane l holds column n=l&15,
// rows m = 8*(l>>4) .. +7 in v8f elements [0..7].
// ---------------------------------------------------------------------------
__device__ __forceinline__ void cinv16(v8f& re, v8f& im) {
  const int l = threadIdx.x & 31;
  const int n = l & 15;
  const int h = l >> 4;
#pragma unroll
  for (int k = 0; k < 16; ++k) {
    const int kh = k >> 3;
    const int kv = k & 7;
    // pivot-row element at this lane's column, and the pivot itself
    float rkr = bperm(re[kv], n | (kh << 4));
    float rki = bperm(im[kv], n | (kh << 4));
    float pr  = bperm(re[kv], k | (kh << 4));
    float pi  = bperm(im[kv], k | (kh << 4));
    float inv = 1.0f / (pr * pr + pi * pi);
    float ipr = pr * inv;
    float ipi = -pi * inv;
    float rksr = rkr * ipr - rki * ipi;   // scaled pivot row
    float rksi = rkr * ipi + rki * ipr;
    const bool pivCol = (n == k);
#pragma unroll
    for (int v = 0; v < 8; ++v) {
      // column-k entry for this lane's row m = 8h+v (pre-update value)
      float fr = bperm(re[v], k | (h << 4));
      float fi = bperm(im[v], k | (h << 4));
      const bool pivRow = (h == kh) && (v == kv);
      float ur  = re[v] - (fr * rksr - fi * rksi);
      float ui  = im[v] - (fr * rksi + fi * rksr);
      float qr  = -(fr * ipr - fi * ipi);
      float qi  = -(fr * ipi + fi * ipr);
      re[v] = pivRow ? (pivCol ? ipr : rksr) : (pivCol ? qr : ur);
      im[v] = pivRow ? (pivCol ? ipi : rksi) : (pivCol ? qi : ui);
    }
  }
}

// ---------------------------------------------------------------------------
// acc-layout <-> LDS row-major 16x16, and global writeback
// ---------------------------------------------------------------------------
__device__ __forceinline__ void store_acc(float* R, float* I, const v8f& cr,
                                          const v8f& ci) {
  const int l = threadIdx.x & 31;
  const int n = l & 15;
  const int mb = (l >> 4) * 8;
#pragma unroll
  for (int v = 0; v < 8; ++v) {
    R[(mb + v) * 16 + n] = cr[v];
    I[(mb + v) * 16 + n] = ci[v];
  }
}

__device__ __forceinline__ void load_acc(const float* R, const float* I,
                                         v8f& cr, v8f& ci) {
  const int l = threadIdx.x & 31;
  const int n = l & 15;
  const int mb = (l >> 4) * 8;
#pragma unroll
  for (int v = 0; v < 8; ++v) {
    cr[v] = R[(mb + v) * 16 + n];
    ci[v] = I[(mb + v) * 16 + n];
  }
}

__device__ __forceinline__ void write_block(float2* __restrict__ out2, int bin,
                                            int ro, int co, const v8f& cr,
                                            const v8f& ci) {
  const int l = threadIdx.x & 31;
  const int n = l & 15;
  const int mb = (l >> 4) * 8;
#pragma unroll
  for (int v = 0; v < 8; ++v) {
    size_t idx = ((size_t)bin * 32 + (size_t)(ro + mb + v)) * 32 + (co + n);
    float2 t;
    t.x = cr[v];
    t.y = ci[v];
    out2[idx] = t;
  }
}

// ---------------------------------------------------------------------------
// Kernel 1: build C = blockdiag(expm(skew(M_g))) coupling matrix (32x32 f32)
// ---------------------------------------------------------------------------
__global__ __launch_bounds__(256) void fdn_setup_kernel(
    const int* __restrict__ delays, const float* __restrict__ gains,
    const float* __restrict__ Mm, const float* __restrict__ alpha,
    float* __restrict__ Cws) {
  __shared__ float T[4][64], R[4][64], W[4][64];
  __shared__ float ph[16];
  __shared__ float gam[32];

  const int t = threadIdx.x;
  const int g = t >> 6, e = t & 63, i = e >> 3, j = e & 7;

  // skew(M), pre-scaled by 2^-8 for scaling-and-squaring
  float x = 0.0f;
  if (j > i)      x =  Mm[g * 64 + i * 8 + j];
  else if (i > j) x = -Mm[g * 64 + j * 8 + i];
  T[g][e] = x * (1.0f / 256.0f);

  if (t < 32) gam[t] = expf((float)delays[t] * logf(gains[t]));

  if (t == 0) {  // Givens product phi (4x4), trivially serial
    float P[16];
    for (int a = 0; a < 16; ++a) P[a] = (a % 5 == 0) ? 1.0f : 0.0f;
    for (int r = 0; r < 3; ++r) {
      float av = alpha[r];
      av = fminf(fmaxf(av, -0.25f * PI_F), 0.25f * PI_F);
      float cc = cosf(av), ss = sinf(av);
      float Q[16];
      for (int rr = 0; rr < 4; ++rr)
        for (int cIdx = 0; cIdx < 4; ++cIdx) {
          float acc = 0.0f;
          for (int kk = 0; kk < 4; ++kk) {
            float cv = (kk == cIdx) ? 1.0f : 0.0f;
            if (kk == r     && cIdx == r)     cv = cc;
            if (kk == r     && cIdx == r + 1) cv = -ss;
            if (kk == r + 1 && cIdx == r)     cv = ss;
            if (kk == r + 1 && cIdx == r + 1) cv = cc;
            acc += P[rr * 4 + kk] * cv;
          }
          Q[rr * 4 + cIdx] = acc;
        }
      for (int a = 0; a < 16; ++a) P[a] = Q[a];
    }
    for (int a = 0; a < 16; ++a) ph[a] = P[a];
  }
  __syncthreads();

  // expm: Horner Taylor (order 6) then 8 squarings; each group of 64 threads
  // owns one 8x8 matrix, one output element per thread.
  R[g][e] = (i == j) ? 1.0f : 0.0f;
  __syncthreads();
  for (int q = 6; q >= 1; --q) {
    float acc = 0.0f;
    for (int kk = 0; kk < 8; ++kk) acc += T[g][i * 8 + kk] * R[g][kk * 8 + j];
    W[g][e] = acc;
    __syncthreads();
    R[g][e] = ((i == j) ? 1.0f : 0.0f) + W[g][e] / (float)q;
    __syncthreads();
  }
  for (int sq = 0; sq < 8; ++sq) {
    float acc = 0.0f;
    for (int kk = 0; kk < 8; ++kk) acc += R[g][i * 8 + kk] * R[g][kk * 8 + j];
    W[g][e] = acc;
    __syncthreads();
    R[g][e] = W[g][e];
    __syncthreads();
  }

  // C[row,col] = phi[g][h] * (U_g @ U_h)[p,q] * gamma[col]
  for (int idx = t; idx < 1024; idx += 256) {
    int row = idx >> 5, col = idx & 31;
    int gg = row >> 3, p = row & 7, hh = col >> 3, qq = col & 7;
    float acc = 0.0f;
    for (int kk = 0; kk < 8; ++kk) acc += R[gg][p * 8 + kk] * R[hh][kk * 8 + qq];
    Cws[idx] = ph[gg * 4 + hh] * acc * gam[col];
  }
}

// ---------------------------------------------------------------------------
// Kernel 2: per-bin inv(D - z*C) via 2x2 block Schur complement,
// one wave32 per bin, WMMA f32 for all 16x16 complex GEMMs.
// ---------------------------------------------------------------------------
__global__ __launch_bounds__(256) void fdn_inv_kernel(
    const float* __restrict__ theta, const int* __restrict__ delays,
    const float* __restrict__ Cws, float2* __restrict__ out2, int K) {
  __shared__ float sC[1024];
  __shared__ int   sD[32];
  __shared__ float sbuf[8][4][2][256];  // wave, buffer, re/im, 16x16

  for (int i = threadIdx.x; i < 1024; i += 256) sC[i] = Cws[i];
  if (threadIdx.x < 32) sD[threadIdx.x] = delays[threadIdx.x];
  __syncthreads();

  const int w = threadIdx.x >> 5;
  const int l = threadIdx.x & 31;
  const int bin = blockIdx.x * 8 + w;
  if (bin >= K) return;  // wave-uniform: EXEC stays all-ones for WMMA

  const int n = l & 15;
  const int h = l >> 4;

  float* b0r = &sbuf[w][0][0][0]; float* b0i = &sbuf[w][0][1][0];  // A^-1
  float* b1r = &sbuf[w][1][0][0]; float* b1i = &sbuf[w][1][1][0];  // B -> A^-1 B
  float* b2r = &sbuf[w][2][0][0]; float* b2i = &sbuf[w][2][1][0];  // C21 -> S^-1
  float* b3r = &sbuf[w][3][0][0]; float* b3i = &sbuf[w][3][1][0];  // CA^-1 -> X21

  const float th = theta[bin];
  float zi, zr;
  sincosf(th, &zi, &zr);  // z = e^{i theta}
  float sdA, cdA, sdD, cdD;
  sincosf(th * (float)sD[n],      &sdA, &cdA);  // z^{d_n}
  sincosf(th * (float)sD[16 + n], &sdD, &cdD);  // z^{d_{16+n}}

  // ---- A = D1 - z*C11 (acc layout, registers), then invert in place ----
  v8f ar, ai;
#pragma unroll
  for (int v = 0; v < 8; ++v) {
    int m = h * 8 + v;
    float c = sC[m * 32 + n];
    ar[v] = ((m == n) ? cdA : 0.0f) - zr * c;
    ai[v] = ((m == n) ? sdA : 0.0f) - zi * c;
  }
  cinv16(ar, ai);
  store_acc(b0r, b0i, ar, ai);

  // ---- off-diagonal blocks: B = -z*C12, C21 = -z*C21 ----
#pragma unroll
  for (int v = 0; v < 8; ++v) {
    int m = h * 8 + v;
    float c12 = sC[m * 32 + 16 + n];
    float c21 = sC[(16 + m) * 32 + n];
    b1r[m * 16 + n] = -zr * c12;  b1i[m * 16 + n] = -zi * c12;
    b2r[m * 16 + n] = -zr * c21;  b2i[m * 16 + n] = -zi * c21;
  }
  lds_fence();

  // ---- AiB = A^-1 @ B ----
  v8f cr = {}, ci = {};
  cgemm(b0r, b0i, b1r, b1i, cr, ci, 1.0f);
  store_acc(b1r, b1i, cr, ci);  // overwrite B
  lds_fence();

  // ---- CAi = C21 @ A^-1 ----
  v8f cr2 = {}, ci2 = {};
  cgemm(b2r, b2i, b0r, b0i, cr2, ci2, 1.0f);
  store_acc(b3r, b3i, cr2, ci2);
  lds_fence();

  // ---- S = D2 - z*C22 - C21 @ AiB, then S^-1 ----
  v8f sr, si;
#pragma unroll
  for (int v = 0; v < 8; ++v) {
    int m = h * 8 + v;
    float c = sC[(16 + m) * 32 + 16 + n];
    sr[v] = ((m == n) ? cdD : 0.0f) - zr * c;
    si[v] = ((m == n) ? sdD : 0.0f) - zi * c;
  }
  cgemm(b2r, b2i, b1r, b1i, sr, si, -1.0f);
  cinv16(sr, si);
  write_block(out2, bin, 16, 16, sr, si);  // X22 = S^-1
  store_acc(b2r, b2i, sr, si);             // overwrite C21 with S^-1
  lds_fence();

  // ---- X21 = -S^-1 @ CAi ----
  v8f xr = {}, xi = {};
  cgemm(b2r, b2i, b3r, b3i, xr, xi, -1.0f);
  write_block(out2, bin, 16, 0, xr, xi);
  store_acc(b3r, b3i, xr, xi);             // overwrite CAi with X21
  lds_fence();

  // ---- X12 = -AiB @ S^-1 ----
  v8f yr = {}, yi = {};
  cgemm(b1r, b1i, b2r, b2i, yr, yi, -1.0f);
  write_block(out2, bin, 0, 16, yr, yi);

  // ---- X11 = A^-1 - AiB @ X21 ----
  v8f wr, wi;
  load_acc(b0r, b0i, wr, wi);
  cgemm(b1r, b1i, b3r, b3i, wr, wi, -1.0f);
  write_block(out2, bin, 0, 0, wr, wi);
}

// ---------------------------------------------------------------------------
extern "C" void kernel_launch(void* const* d_in, const int* in_sizes, int n_in,
                              void* d_out, int out_size, void* d_ws,
                              size_t ws_size, hipStream_t stream) {
  const float* theta  = (const float*)d_in[0];
  const int*   delays = (const int*)d_in[1];
  const float* gains  = (const float*)d_in[2];
  const float* Mm     = (const float*)d_in[3];
  const float* alpha  = (const float*)d_in[4];
  float* Cws = (float*)d_ws;  // 32*32 floats of scratch
  const int K = in_sizes[0];

  fdn_setup_kernel<<<1, 256, 0, stream>>>(delays, gains, Mm, alpha, Cws);

  const int nblocks = (K + 7) / 8;  // 8 bins (waves) per 256-thread block
  fdn_inv_kernel<<<nblocks, 256, 0, stream>>>(theta, delays, Cws,
                                              (float2*)d_out, K);
}